// StableHierarchicalPooling_29274497089982
// MI455X (gfx1250) — compile-verified
//
#include <hip/hip_runtime.h>
#include <hip/hip_bf16.h>
#include <math.h>

// ---------------- problem constants ----------------
#define N_NODES 100000
#define C_DIM   128
#define K_SN    32
#define B_GR    16
#define EPS     1e-9f
#define TILES   6250                 // N_NODES / 16 (exact)

// ---------------- d_out layout (floats) ----------------
#define OUT_SZ  (B_GR * K_SN * C_DIM)   // 65536
#define S_OFF   OUT_SZ
#define S_SZ    (N_NODES * K_SN)        // 3,200,000
#define MU_OFF  (S_OFF + S_SZ)
#define MU_SZ   (B_GR * K_SN * 2)       // 1024
#define LOSS_OFF (MU_OFF + MU_SZ)

// ---------------- ws layout ----------------
#define WS_COLSUM 0      // 32 floats: sum_i s[i,k]
#define WS_CPOS   32     // 64 floats: sum_i s[i,k]*pos[i,{x,y}]
#define WS_CPSQ   96     // 32 floats: sum_i s[i,k]*|pos_i|^2
#define WS_ENT    128    // 1 float : sum_i sum_k s*log(s+eps)
#define WS_BS     160    // 512 floats: per-graph sum_i s[i,k]
#define WS_BP     672    // 1024 floats: per-graph sum_i s[i,k]*pos
#define WS_FLOATS 1696
#define WS_STARTS_BYTE 8192   // (B+1) ints

typedef __attribute__((ext_vector_type(2))) float v2f;
typedef __attribute__((ext_vector_type(8))) float v8f;

static __device__ __forceinline__ v8f v8f_zero() {
  v8f z;
#pragma unroll
  for (int i = 0; i < 8; ++i) z[i] = 0.0f;
  return z;
}

static __device__ __forceinline__ v8f wmma4(v2f a, v2f b, v8f c) {
  // D = A(16x4 f32) * B(4x16 f32) + C(16x16 f32)
  return __builtin_amdgcn_wmma_f32_16x16x4_f32(false, a, false, b,
                                               (short)0, c, false, false);
}

// ---------------- kernel 0: zero init ----------------
__global__ void init_kernel(float* __restrict__ out, float* __restrict__ wsf) {
  int i = blockIdx.x * blockDim.x + threadIdx.x;
  const int total = OUT_SZ + WS_FLOATS;
  for (; i < total; i += gridDim.x * blockDim.x) {
    if (i < OUT_SZ) out[i] = 0.0f;
    else            wsf[i - OUT_SZ] = 0.0f;
  }
}

// ---------------- kernel 1: segment boundaries (batch is sorted) ----------------
__global__ void seg_starts_kernel(const int* __restrict__ batch,
                                  int* __restrict__ starts) {
  int i = blockIdx.x * blockDim.x + threadIdx.x;
  if (i >= N_NODES) return;
  int bi = batch[i];
  if (i == 0) {
    for (int b = 0; b <= bi; ++b) starts[b] = 0;
  } else {
    int bp = batch[i - 1];
    if (bi != bp) for (int b = bp + 1; b <= bi; ++b) starts[b] = i;
  }
  if (i == N_NODES - 1) {
    for (int b = bi + 1; b <= B_GR; ++b) starts[b] = N_NODES;
  }
}

// ---------------- kernel 2: fused MLP + gumbel-softmax + reductions ----------------
// 4 waves / block, one 16-node tile per wave.
__global__ __launch_bounds__(128) void assign_kernel(
    const float* __restrict__ x,   const int* __restrict__ batch,
    const float* __restrict__ pos, const float* __restrict__ gu,
    const float* __restrict__ W1,  const float* __restrict__ b1,
    const float* __restrict__ W2,  const float* __restrict__ b2,
    const float* __restrict__ scaling_p, const float* __restrict__ amask,
    float* __restrict__ s_out, float* __restrict__ wsf)
{
  __shared__ float lds_h[4][16][132];   // h tile per wave, padded stride
  __shared__ float lds_lg[4][16][33];   // logits tile per wave
  __shared__ float s_tile[64][32];      // s for 64 nodes of this block
  __shared__ float lds_pos[64][2];
  __shared__ int   lds_b[64];
  __shared__ float lds_ent[64];

  const int tid   = threadIdx.x;
  const int wave  = tid >> 5;
  const int lane  = tid & 31;
  const int tile  = blockIdx.x * 4 + wave;
  const int n_lo  = lane & 15;           // N (col) / M (A-row) index
  const int khalf = (lane >> 4) << 1;    // 0 or 2: K pair per lane half
  const int rhalf = (lane >> 4) << 3;    // 0 or 8: D row offset per half
  const float scaling = scaling_p[0];

  if (tile < TILES) {                    // wave-uniform guard: EXEC all-ones
    const int row0 = tile * 16;
    // ---- layer 1: h = relu(x @ W1 + b1), 16x128 tile via WMMA ----
    v8f acc[8];
#pragma unroll
    for (int cb = 0; cb < 8; ++cb) acc[cb] = v8f_zero();
    const float* xrow = x + (size_t)(row0 + n_lo) * C_DIM;
    for (int k0 = 0; k0 < C_DIM; k0 += 4) {
      const v2f a = *reinterpret_cast<const v2f*>(xrow + k0 + khalf); // b64 load
#pragma unroll
      for (int cb = 0; cb < 8; ++cb) {
        v2f b;
        b.x = W1[(size_t)(k0 + khalf)     * C_DIM + cb * 16 + n_lo];
        b.y = W1[(size_t)(k0 + khalf + 1) * C_DIM + cb * 16 + n_lo];
        acc[cb] = wmma4(a, b, acc[cb]);
      }
    }
#pragma unroll
    for (int cb = 0; cb < 8; ++cb) {
      const float bias = b1[cb * 16 + n_lo];
#pragma unroll
      for (int r = 0; r < 8; ++r) {
        float hv = acc[cb][r] + bias;
        lds_h[wave][r + rhalf][cb * 16 + n_lo] = hv > 0.0f ? hv : 0.0f;
      }
    }
    // ---- layer 2: logits = (h @ W2 + b2) * scaling ----
    v8f acc2[2];
    acc2[0] = v8f_zero(); acc2[1] = v8f_zero();
    for (int k0 = 0; k0 < C_DIM; k0 += 4) {
      const v2f a = *reinterpret_cast<const v2f*>(&lds_h[wave][n_lo][k0 + khalf]);
#pragma unroll
      for (int cb = 0; cb < 2; ++cb) {
        v2f b;
        b.x = W2[(size_t)(k0 + khalf)     * K_SN + cb * 16 + n_lo];
        b.y = W2[(size_t)(k0 + khalf + 1) * K_SN + cb * 16 + n_lo];
        acc2[cb] = wmma4(a, b, acc2[cb]);
      }
    }
#pragma unroll
    for (int cb = 0; cb < 2; ++cb) {
      const int col = cb * 16 + n_lo;
      const float bb = b2[col];
      const float mk = amask[col];
#pragma unroll
      for (int r = 0; r < 8; ++r) {
        float lg = (acc2[cb][r] + bb) * scaling;
        if (mk == 0.0f) lg = -1e9f;
        lds_lg[wave][r + rhalf][col] = lg;
      }
    }
    // ---- gumbel-softmax: one lane per node row ----
    if (lane < 16) {
      const int node = row0 + lane;
      const float* gurow = gu + (size_t)node * K_SN;
      float z[K_SN];
      float zmax = -3.4e38f;
#pragma unroll
      for (int k = 0; k < K_SN; ++k) {
        float u = gurow[k];
        float g = -logf(-logf(u + EPS) + EPS);
        float zz = lds_lg[wave][lane][k] + g;   // TAU = 1
        z[k] = zz;
        zmax = fmaxf(zmax, zz);
      }
      float sum = 0.0f;
#pragma unroll
      for (int k = 0; k < K_SN; ++k) { float e = expf(z[k] - zmax); z[k] = e; sum += e; }
      const float inv = 1.0f / sum;
      float ent = 0.0f;
      float* so = s_out + (size_t)node * K_SN;
#pragma unroll
      for (int k = 0; k < K_SN; ++k) {
        float sv = z[k] * inv;
        s_tile[wave * 16 + lane][k] = sv;
        so[k] = sv;
        ent += sv * logf(sv + EPS);
      }
      lds_ent[wave * 16 + lane] = ent;
    }
  }

  // ---- stage pos/batch for the block's 64 nodes ----
  const int base  = blockIdx.x * 64;
  const int nrows = (N_NODES - base) < 64 ? (N_NODES - base) : 64;
  if (tid < 64 && tid < nrows) {
    const int node = base + tid;
    lds_pos[tid][0] = pos[(size_t)node * 2];
    lds_pos[tid][1] = pos[(size_t)node * 2 + 1];
    lds_b[tid] = batch[node];
  }
  __syncthreads();

  // ---- block reduction: one thread per super-node column ----
  if (tid < K_SN) {
    const int k = tid;
    float cs = 0, cpx = 0, cpy = 0, cq = 0;
    int curb = -1; float bs = 0, bpx = 0, bpy = 0;
    for (int r = 0; r < nrows; ++r) {
      const float sv = s_tile[r][k];
      const float px = lds_pos[r][0], py = lds_pos[r][1];
      cs += sv; cpx += sv * px; cpy += sv * py; cq += sv * (px * px + py * py);
      const int b = lds_b[r];
      if (b != curb) {
        if (curb >= 0) {
          atomicAdd(&wsf[WS_BS + curb * K_SN + k], bs);
          atomicAdd(&wsf[WS_BP + (curb * K_SN + k) * 2],     bpx);
          atomicAdd(&wsf[WS_BP + (curb * K_SN + k) * 2 + 1], bpy);
        }
        curb = b; bs = 0; bpx = 0; bpy = 0;
      }
      bs += sv; bpx += sv * px; bpy += sv * py;
    }
    if (curb >= 0) {
      atomicAdd(&wsf[WS_BS + curb * K_SN + k], bs);
      atomicAdd(&wsf[WS_BP + (curb * K_SN + k) * 2],     bpx);
      atomicAdd(&wsf[WS_BP + (curb * K_SN + k) * 2 + 1], bpy);
    }
    atomicAdd(&wsf[WS_COLSUM + k], cs);
    atomicAdd(&wsf[WS_CPOS + 2 * k],     cpx);
    atomicAdd(&wsf[WS_CPOS + 2 * k + 1], cpy);
    atomicAdd(&wsf[WS_CPSQ + k], cq);
  } else if (tid == K_SN) {
    float e = 0.0f;
    for (int r = 0; r < nrows; ++r) e += lds_ent[r];
    atomicAdd(&wsf[WS_ENT], e);
  }
}

// ---------------- kernel 3: scalar losses + centroids + separation ----------------
__global__ __launch_bounds__(256) void finalize_kernel(
    const float* __restrict__ wsf, const float* __restrict__ amask,
    float* __restrict__ out)
{
  __shared__ float sh_avg[K_SN], sh_cs[K_SN];
  __shared__ float sh_mu[B_GR * K_SN * 2];
  __shared__ float sh_red[256];
  const int t = threadIdx.x;
  if (t < K_SN) {
    const float cs = wsf[WS_COLSUM + t];
    sh_cs[t] = cs;
    sh_avg[t] = cs / (float)N_NODES;
  }
  for (int e = t; e < B_GR * K_SN; e += 256) {
    const float den = wsf[WS_BS + e] + EPS;
    const float mx = wsf[WS_BP + 2 * e]     / den;
    const float my = wsf[WS_BP + 2 * e + 1] / den;
    sh_mu[2 * e] = mx; sh_mu[2 * e + 1] = my;
    out[MU_OFF + 2 * e] = mx; out[MU_OFF + 2 * e + 1] = my;
  }
  __syncthreads();
  float sep = 0.0f;
  for (int p = t; p < B_GR * K_SN * K_SN; p += 256) {
    const int b = p / (K_SN * K_SN);
    const int k = (p / K_SN) % K_SN;
    const int j = p % K_SN;
    if (j != k) {
      const float dx = sh_mu[(b * K_SN + k) * 2]     - sh_mu[(b * K_SN + j) * 2];
      const float dy = sh_mu[(b * K_SN + k) * 2 + 1] - sh_mu[(b * K_SN + j) * 2 + 1];
      sep += 1.0f / (dx * dx + dy * dy + 1.0f);
    }
  }
  sh_red[t] = sep;
  __syncthreads();
  for (int off = 128; off > 0; off >>= 1) {
    if (t < off) sh_red[t] += sh_red[t + off];
    __syncthreads();
  }
  if (t == 0) {
    const float entropy = -wsf[WS_ENT] / (float)N_NODES;
    const float logK = logf((float)K_SN);
    const float up = 1.0f / (float)K_SN;
    float div = 0, ent_avg = 0, bal = 0, mx = 0, prun = 0, msum = 0, spat = 0;
    for (int k = 0; k < K_SN; ++k) {
      const float av = sh_avg[k];
      div     += up * logf(up / (av + EPS));
      ent_avg -= av * logf(av + EPS);
      bal     += av * logf(av * (float)K_SN + EPS);
      mx       = fmaxf(mx, av);
      prun    += fabsf(av * (1.0f - amask[k]));
      msum    += amask[k];
      const float ssum = sh_cs[k] + EPS;
      const float mux = wsf[WS_CPOS + 2 * k]     / ssum;
      const float muy = wsf[WS_CPOS + 2 * k + 1] / ssum;
      spat    += wsf[WS_CPSQ + k] / ssum - mux * mux - muy * muy;
    }
    prun /= (float)K_SN;
    spat /= (float)K_SN;
    const float sparsity = msum / (float)K_SN * 0.01f;
    const float collapse = ((logK - ent_avg) / logK +
                            fmaxf(mx - 2.0f / (float)K_SN, 0.0f)) * 2.0f;
    const float separ = sh_red[0] / ((float)(K_SN * (K_SN - 1)) + EPS);
    out[LOSS_OFF + 0] = entropy;
    out[LOSS_OFF + 1] = div;
    out[LOSS_OFF + 2] = spat;
    out[LOSS_OFF + 3] = prun;
    out[LOSS_OFF + 4] = sparsity;
    out[LOSS_OFF + 5] = 0.0f;       // temporal * TEMPORAL_W
    out[LOSS_OFF + 6] = collapse;
    out[LOSS_OFF + 7] = bal;
    out[LOSS_OFF + 8] = separ;
  }
}

// ---------------- kernel 4: out[b] = s_b^T @ x_b via WMMA ----------------
// One wave computes the FULL 32x128 output tile for its node chunk:
// 16 accumulator fragments (128 VGPRs). x and s are each read exactly once.
#define POOL_NS 16   // node-range splits per graph
__global__ __launch_bounds__(32) void pool_kernel(
    const float* __restrict__ x, const float* __restrict__ s,
    const int* __restrict__ starts, float* __restrict__ out)
{
  const int lane = threadIdx.x;
  int bid = blockIdx.x;
  const int sp = bid % POOL_NS;
  const int b  = bid / POOL_NS;
  const int i0 = starts[b], i1 = starts[b + 1];
  const int cnt = i1 - i0;
  if (cnt <= 0) return;
  const int chunk = (cnt + POOL_NS - 1) / POOL_NS;
  const int c0 = i0 + sp * chunk;
  const int c1 = (c0 + chunk) < i1 ? (c0 + chunk) : i1;
  if (c0 >= c1) return;

  const int n_lo  = lane & 15;
  const int khalf = (lane >> 4) << 1;
  const int rhalf = (lane >> 4) << 3;

  v8f acc[16];                       // acc[kb*8+cb]
#pragma unroll
  for (int t = 0; t < 16; ++t) acc[t] = v8f_zero();

  const int main_end = c0 + ((c1 - c0) & ~3);
  for (int i = c0; i < main_end; i += 4) {
    const size_t r0 = (size_t)(i + khalf);
    const size_t r1 = r0 + 1;
    v2f a0, a1;                      // A fragments for k-blocks 0 and 1
    a0.x = s[r0 * K_SN + n_lo];       a0.y = s[r1 * K_SN + n_lo];
    a1.x = s[r0 * K_SN + 16 + n_lo];  a1.y = s[r1 * K_SN + 16 + n_lo];
    v2f bx[8];                       // B fragments for all 8 c-blocks
#pragma unroll
    for (int cb = 0; cb < 8; ++cb) {
      bx[cb].x = x[r0 * C_DIM + cb * 16 + n_lo];
      bx[cb].y = x[r1 * C_DIM + cb * 16 + n_lo];
    }
#pragma unroll
    for (int cb = 0; cb < 8; ++cb) {
      acc[cb]     = wmma4(a0, bx[cb], acc[cb]);
      acc[8 + cb] = wmma4(a1, bx[cb], acc[8 + cb]);
    }
  }
  if (main_end < c1) {               // uniform tail; per-lane zero-fill keeps EXEC full
    const int ia = main_end + khalf;
    const int ib = ia + 1;
    const bool va = ia < c1, vb = ib < c1;
    v2f a0, a1;
    a0.x = va ? s[(size_t)ia * K_SN + n_lo]      : 0.0f;
    a0.y = vb ? s[(size_t)ib * K_SN + n_lo]      : 0.0f;
    a1.x = va ? s[(size_t)ia * K_SN + 16 + n_lo] : 0.0f;
    a1.y = vb ? s[(size_t)ib * K_SN + 16 + n_lo] : 0.0f;
    v2f bx[8];
#pragma unroll
    for (int cb = 0; cb < 8; ++cb) {
      bx[cb].x = va ? x[(size_t)ia * C_DIM + cb * 16 + n_lo] : 0.0f;
      bx[cb].y = vb ? x[(size_t)ib * C_DIM + cb * 16 + n_lo] : 0.0f;
    }
#pragma unroll
    for (int cb = 0; cb < 8; ++cb) {
      acc[cb]     = wmma4(a0, bx[cb], acc[cb]);
      acc[8 + cb] = wmma4(a1, bx[cb], acc[8 + cb]);
    }
  }
#pragma unroll
  for (int kb = 0; kb < 2; ++kb) {
#pragma unroll
    for (int cb = 0; cb < 8; ++cb) {
#pragma unroll
      for (int r = 0; r < 8; ++r) {
        const int kk = kb * 16 + r + rhalf;
        const int cc = cb * 16 + n_lo;
        atomicAdd(&out[((size_t)b * K_SN + kk) * C_DIM + cc], acc[kb * 8 + cb][r]);
      }
    }
  }
}

// ---------------- launch ----------------
extern "C" void kernel_launch(void* const* d_in, const int* in_sizes, int n_in,
                              void* d_out, int out_size, void* d_ws, size_t ws_size,
                              hipStream_t stream) {
  (void)in_sizes; (void)n_in; (void)out_size; (void)ws_size;
  const float* x     = (const float*)d_in[0];
  const int*   batch = (const int*)  d_in[1];
  const float* pos   = (const float*)d_in[2];
  const float* gu    = (const float*)d_in[3];
  const float* W1    = (const float*)d_in[4];
  const float* b1    = (const float*)d_in[5];
  const float* W2    = (const float*)d_in[6];
  const float* b2    = (const float*)d_in[7];
  const float* scal  = (const float*)d_in[8];
  const float* amask = (const float*)d_in[9];
  float* out = (float*)d_out;
  float* wsf = (float*)d_ws;
  int* starts = (int*)((char*)d_ws + WS_STARTS_BYTE);

  init_kernel<<<264, 256, 0, stream>>>(out, wsf);
  seg_starts_kernel<<<(N_NODES + 255) / 256, 256, 0, stream>>>(batch, starts);
  assign_kernel<<<(TILES + 3) / 4, 128, 0, stream>>>(
      x, batch, pos, gu, W1, b1, W2, b2, scal, amask, out + S_OFF, wsf);
  finalize_kernel<<<1, 256, 0, stream>>>(wsf, amask, out);
  pool_kernel<<<B_GR * POOL_NS, 32, 0, stream>>>(
      x, out + S_OFF, starts, out);
}